// GAT_73933567033796
// MI455X (gfx1250) — compile-verified
//
#include <hip/hip_runtime.h>

#define NUM_NODES 50000
#define NUM_EDGES 800000
#define IN_FEATS  10
#define HIDDEN    128
#define E_TOTAL   (NUM_EDGES + NUM_NODES)
#define NEG_ATT   0.2f
#define NEG_ACT   0.01f

typedef __attribute__((ext_vector_type(2))) float v2f;
typedef __attribute__((ext_vector_type(8))) float v8f;

__device__ __forceinline__ float lrelu(float v, float s) {
    return v > 0.0f ? v : s * v;
}

// float atomic max via signed-max / unsigned-min trick (init with -inf)
__device__ __forceinline__ void atomicMaxFloat(float* addr, float val) {
    if (val >= 0.0f)
        atomicMax((int*)addr, __float_as_int(val));
    else
        atomicMin((unsigned int*)addr, __float_as_uint(val));
}

// ---------------- Layer-1 transform: h = x @ W1, alpha_src/dst = h @ a ----------------
__global__ void __launch_bounds__(HIDDEN)
l1_transform(const float* __restrict__ x, const float* __restrict__ W1,
             const float* __restrict__ a_src, const float* __restrict__ a_dst,
             float* __restrict__ h, float* __restrict__ as, float* __restrict__ ad) {
    int n = blockIdx.x;
    int c = threadIdx.x;            // 0..127
    __shared__ float xs[IN_FEATS];
    __shared__ float rs[HIDDEN];
    __shared__ float rd[HIDDEN];
    if (c < IN_FEATS) xs[c] = x[n * IN_FEATS + c];
    __syncthreads();
    float acc = 0.0f;
#pragma unroll
    for (int k = 0; k < IN_FEATS; ++k)
        acc = fmaf(xs[k], W1[k * HIDDEN + c], acc);
    h[(size_t)n * HIDDEN + c] = acc;
    rs[c] = acc * a_src[c];
    rd[c] = acc * a_dst[c];
    __syncthreads();
    for (int off = HIDDEN / 2; off > 0; off >>= 1) {
        if (c < off) { rs[c] += rs[c + off]; rd[c] += rd[c + off]; }
        __syncthreads();
    }
    if (c == 0) { as[n] = rs[0]; ad[n] = rd[0]; }
}

// ---------------- Wave-per-node attention dots (layer 2) ----------------
__global__ void __launch_bounds__(256)
node_alpha(const float* __restrict__ h, const float* __restrict__ a_src,
           const float* __restrict__ a_dst, float* __restrict__ as, float* __restrict__ ad) {
    int wave = threadIdx.x >> 5;
    int lane = threadIdx.x & 31;
    int n = blockIdx.x * 8 + wave;
    if (n >= NUM_NODES) return;
    const float* row = h + (size_t)n * HIDDEN;
    float s = 0.0f, d = 0.0f;
#pragma unroll
    for (int q = 0; q < HIDDEN / 32; ++q) {
        int i = lane + q * 32;
        float v = row[i];
        s = fmaf(v, a_src[i], s);
        d = fmaf(v, a_dst[i], d);
    }
#pragma unroll
    for (int off = 16; off > 0; off >>= 1) {
        s += __shfl_xor(s, off, 32);
        d += __shfl_xor(d, off, 32);
    }
    if (lane == 0) { as[n] = s; ad[n] = d; }
}

// ---------------- per-layer init: m=-inf, z=0, agg=0 ----------------
__global__ void init_layer(float* __restrict__ m, float* __restrict__ z,
                           float* __restrict__ agg) {
    int i = blockIdx.x * blockDim.x + threadIdx.x;
    if (i < NUM_NODES) { m[i] = __int_as_float(0xFF800000); z[i] = 0.0f; }
    if (i < NUM_NODES * HIDDEN) agg[i] = 0.0f;
}

__device__ __forceinline__ void edge_endpoints(const int* __restrict__ ei, int e,
                                               int& s, int& d) {
    if (e < NUM_EDGES) { s = ei[e]; d = ei[NUM_EDGES + e]; }
    else               { s = d = e - NUM_EDGES; }   // self loops
}

// ---------------- pass 1: segment max of attention logits ----------------
__global__ void edge_max(const int* __restrict__ ei, const float* __restrict__ as,
                         const float* __restrict__ ad, float* __restrict__ m) {
    int e = blockIdx.x * blockDim.x + threadIdx.x;
    if (e >= E_TOTAL) return;
    int s, d; edge_endpoints(ei, e, s, d);
    float v = lrelu(as[s] + ad[d], NEG_ATT);
    atomicMaxFloat(&m[d], v);
}

// ---------------- pass 2: w = exp(e - m[dst]); z = segment sum ----------------
__global__ void edge_expsum(const int* __restrict__ ei, const float* __restrict__ as,
                            const float* __restrict__ ad, const float* __restrict__ m,
                            float* __restrict__ w, float* __restrict__ z) {
    int e = blockIdx.x * blockDim.x + threadIdx.x;
    if (e >= E_TOTAL) return;
    int s, d; edge_endpoints(ei, e, s, d);
    float v = lrelu(as[s] + ad[d], NEG_ATT);
    float ww = expf(v - m[d]);
    w[e] = ww;
    atomicAdd(&z[d], ww);
}

// ---------------- pass 3: wave-per-edge scatter aggregate ----------------
__global__ void __launch_bounds__(256)
edge_aggregate(const int* __restrict__ ei, const float* __restrict__ w,
               const float* __restrict__ z, const float* __restrict__ h,
               float* __restrict__ agg) {
    int wave = threadIdx.x >> 5;
    int lane = threadIdx.x & 31;
    int e = blockIdx.x * 8 + wave;
    if (e >= E_TOTAL) return;
    int s, d; edge_endpoints(ei, e, s, d);
    float coeff = w[e] / (z[d] + 1e-16f);
    const float* hs = h + (size_t)s * HIDDEN;
    float*       od = agg + (size_t)d * HIDDEN;
#pragma unroll
    for (int q = 0; q < HIDDEN / 32; ++q) {
        int c = lane + q * 32;                 // coalesced gather + coalesced atomics
        atomicAdd(&od[c], hs[c] * coeff);
    }
}

// ---------------- finalize: out = leaky_relu(agg + bias) ----------------
__global__ void finalize(const float* __restrict__ agg, const float* __restrict__ bias,
                         float* __restrict__ out) {
    int idx = blockIdx.x * blockDim.x + threadIdx.x;
    if (idx >= NUM_NODES * HIDDEN) return;
    float v = agg[idx] + bias[idx & (HIDDEN - 1)];
    out[idx] = lrelu(v, NEG_ACT);
}

// ---------------- Layer-2 GEMM via V_WMMA_F32_16X16X4_F32 ----------------
// C[N,128] = A[N,128] @ B[128,128], 8 waves/block, wave w -> 16x16 tile at (blockIdx*16, w*16)
__global__ void __launch_bounds__(256)
gemm_wmma(const float* __restrict__ A, const float* __restrict__ B, float* __restrict__ C) {
    __shared__ float As[16 * HIDDEN];          // 8 KB A tile, staged once per block
    const int m0 = blockIdx.x * 16;
    const int tid = threadIdx.x;
    {
        const float4* Ag  = (const float4*)(A + (size_t)m0 * HIDDEN);
        float4*       Asv = (float4*)As;
        for (int i = tid; i < 16 * HIDDEN / 4; i += 256) Asv[i] = Ag[i];
    }
    __syncthreads();

    const int wave = tid >> 5;
    const int lane = tid & 31;
    const int n0   = wave * 16;
    const int half = lane >> 4;    // 0: K lo pair, 1: K hi pair
    const int l    = lane & 15;

    v8f acc = {};
    for (int k0 = 0; k0 < HIDDEN; k0 += 4) {
        // A frag: row l, cols (k0 + 2*half + {0,1})
        const float* ap = &As[l * HIDDEN + k0 + 2 * half];
        v2f a; a.x = ap[0]; a.y = ap[1];
        // B frag: rows (k0 + 2*half + {0,1}), col n0 + l
        const float* bp = B + (size_t)(k0 + 2 * half) * HIDDEN + n0 + l;
        v2f b; b.x = bp[0]; b.y = bp[HIDDEN];
        acc = __builtin_amdgcn_wmma_f32_16x16x4_f32(
            false, a, false, b, (short)0, acc, false, false);
    }
    // D layout: VGPR r -> row m0 + r + 8*half, col n0 + l
    float* crow = C + (size_t)(m0 + 8 * half) * HIDDEN + n0 + l;
#pragma unroll
    for (int r = 0; r < 8; ++r) crow[(size_t)r * HIDDEN] = acc[r];
}

extern "C" void kernel_launch(void* const* d_in, const int* in_sizes, int n_in,
                              void* d_out, int out_size, void* d_ws, size_t ws_size,
                              hipStream_t stream) {
    const float* x      = (const float*)d_in[0];
    const int*   ei     = (const int*)  d_in[1];   // [2, NUM_EDGES]
    const float* W1     = (const float*)d_in[2];
    const float* a_src1 = (const float*)d_in[3];
    const float* a_dst1 = (const float*)d_in[4];
    const float* b1     = (const float*)d_in[5];
    const float* W2     = (const float*)d_in[6];
    const float* a_src2 = (const float*)d_in[7];
    const float* a_dst2 = (const float*)d_in[8];
    const float* b2     = (const float*)d_in[9];
    float* out = (float*)d_out;

    float* ws   = (float*)d_ws;
    const size_t NH = (size_t)NUM_NODES * HIDDEN;
    float* hbuf = ws;               // N*128
    float* act  = hbuf + NH;        // N*128
    float* agg  = act  + NH;        // N*128
    float* as   = agg  + NH;        // N
    float* ad   = as   + NUM_NODES; // N
    float* m    = ad   + NUM_NODES; // N
    float* z    = m    + NUM_NODES; // N
    float* wbuf = z    + NUM_NODES; // E_TOTAL

    const int NHgrid = (int)((NH + 255) / 256);
    const int Egrid  = (E_TOTAL + 255) / 256;
    const int Ewave  = (E_TOTAL + 7) / 8;

    // ---- Layer 1 ----
    l1_transform<<<NUM_NODES, HIDDEN, 0, stream>>>(x, W1, a_src1, a_dst1, hbuf, as, ad);
    init_layer<<<NHgrid, 256, 0, stream>>>(m, z, agg);
    edge_max<<<Egrid, 256, 0, stream>>>(ei, as, ad, m);
    edge_expsum<<<Egrid, 256, 0, stream>>>(ei, as, ad, m, wbuf, z);
    edge_aggregate<<<Ewave, 256, 0, stream>>>(ei, wbuf, z, hbuf, agg);
    finalize<<<NHgrid, 256, 0, stream>>>(agg, b1, act);

    // ---- Layer 2 ----
    gemm_wmma<<<NUM_NODES / 16, 256, 0, stream>>>(act, W2, hbuf);
    node_alpha<<<(NUM_NODES + 7) / 8, 256, 0, stream>>>(hbuf, a_src2, a_dst2, as, ad);
    init_layer<<<NHgrid, 256, 0, stream>>>(m, z, agg);
    edge_max<<<Egrid, 256, 0, stream>>>(ei, as, ad, m);
    edge_expsum<<<Egrid, 256, 0, stream>>>(ei, as, ad, m, wbuf, z);
    edge_aggregate<<<Ewave, 256, 0, stream>>>(ei, wbuf, z, hbuf, agg);
    finalize<<<NHgrid, 256, 0, stream>>>(agg, b2, out);
}